// MotionLoss_53936199303671
// MI455X (gfx1250) — compile-verified
//
#include <hip/hip_runtime.h>
#include <math.h>

typedef __attribute__((ext_vector_type(2))) float v2f;
typedef __attribute__((ext_vector_type(8))) float v8f;

namespace {
constexpr int NJ  = 22;
constexpr int ROW = NJ * 6;            // 132 floats per (b,s) sample
constexpr int TB  = 64;                // threads per block == frames per chunk
constexpr int POSSTRIDE = 133;         // 132 padded -> odd stride, bank-conflict free
constexpr int SDIM = 512;
constexpr int BDIM = 256;
constexpr int NCHUNK = SDIM / TB;      // 8
constexpr int NBLK   = BDIM * NCHUNK;  // 2048
constexpr int PSTRIDE = 16;            // partial slots per block in workspace
constexpr int ACCSTRIDE = 17;          // padded per-thread accumulator row in LDS
}

// cont6d -> rotation matrix, column-major: R[0..2]=b1, R[3..5]=b2, R[6..8]=b3
__device__ __forceinline__ void d6_to_mat(const float* __restrict__ d6, float R[9]) {
  float a1x = d6[0], a1y = d6[1], a1z = d6[2];
  float a2x = d6[3], a2y = d6[4], a2z = d6[5];
  float inv1 = 1.0f / fmaxf(sqrtf(a1x*a1x + a1y*a1y + a1z*a1z), 1e-12f);
  float b1x = a1x*inv1, b1y = a1y*inv1, b1z = a1z*inv1;
  float dp = b1x*a2x + b1y*a2y + b1z*a2z;
  float ux = a2x - b1x*dp, uy = a2y - b1y*dp, uz = a2z - b1z*dp;
  float inv2 = 1.0f / fmaxf(sqrtf(ux*ux + uy*uy + uz*uz), 1e-12f);
  float b2x = ux*inv2, b2y = uy*inv2, b2z = uz*inv2;
  R[0]=b1x; R[1]=b1y; R[2]=b1z;
  R[3]=b2x; R[4]=b2y; R[5]=b2z;
  R[6] = b1y*b2z - b1z*b2y;          // b3 = b1 x b2
  R[7] = b1z*b2x - b1x*b2z;
  R[8] = b1x*b2y - b1y*b2x;
}

// Full 22-joint FK for one frame; DFS over the skeleton tree keeping only a
// 2-deep save stack (branch points at joints 1 and 3) -> ~36 live floats.
// Writes 66 floats (22 joints x xyz) of global positions into LDS.
__device__ void fk_full(const float* __restrict__ row,
                        const float* __restrict__ offs,
                        float* __restrict__ outPos) {
  float g[9] = {1.f,0.f,0.f, 0.f,1.f,0.f, 0.f,0.f,1.f};
  float p[3] = {0.f,0.f,0.f};
  outPos[0]=0.f; outPos[1]=0.f; outPos[2]=0.f;
  float gA[9], pA[3], gB[9], pB[3];

  auto step = [&](int j) {
    float R[9];
    d6_to_mat(row + j*6, R);
    float ng[9];
#pragma unroll
    for (int c = 0; c < 3; ++c) {
#pragma unroll
      for (int r = 0; r < 3; ++r) {
        ng[3*c + r] = g[r]*R[3*c] + g[3 + r]*R[3*c + 1] + g[6 + r]*R[3*c + 2];
      }
    }
    float ox = offs[j*3], oy = offs[j*3+1], oz = offs[j*3+2];
    float npx = p[0] + g[0]*ox + g[3]*oy + g[6]*oz;
    float npy = p[1] + g[1]*ox + g[4]*oy + g[7]*oz;
    float npz = p[2] + g[2]*ox + g[5]*oy + g[8]*oz;
#pragma unroll
    for (int k = 0; k < 9; ++k) g[k] = ng[k];
    p[0]=npx; p[1]=npy; p[2]=npz;
    outPos[j*3]=npx; outPos[j*3+1]=npy; outPos[j*3+2]=npz;
  };
  auto save = [&](float* gs, float* ps){
#pragma unroll
    for (int k=0;k<9;++k) gs[k]=g[k];
    ps[0]=p[0]; ps[1]=p[1]; ps[2]=p[2];
  };
  auto restore = [&](const float* gs, const float* ps){
#pragma unroll
    for (int k=0;k<9;++k) g[k]=gs[k];
    p[0]=ps[0]; p[1]=ps[1]; p[2]=ps[2];
  };

  // PARENTS = [-1,0,1,2,3,4,3,6,7,8,3,10,11,12,1,14,15,16,17,1,19,20]
  step(1);
  save(gA, pA);                                    // joint 1 (branch: 2,14,19)
  step(2); step(3);
  save(gB, pB);                                    // joint 3 (branch: 4,6,10)
  step(4); step(5);
  restore(gB, pB); step(6);  step(7);  step(8);  step(9);
  restore(gB, pB); step(10); step(11); step(12); step(13);
  restore(gA, pA); step(14); step(15); step(16); step(17); step(18);
  restore(gA, pA); step(19); step(20); step(21);
}

__global__ __launch_bounds__(TB) void motion_loss_main(
    const float* __restrict__ pred, const float* __restrict__ tgt,
    const float* __restrict__ offsets, float* __restrict__ partials) {
  __shared__ float offs[NJ * 3];
  __shared__ float pos[(TB + 2) * POSSTRIDE];      // per row: [0..65]=pred, [66..131]=gt
  __shared__ float accL[TB * ACCSTRIDE];           // per-thread 16 accumulator slots

  const int t   = threadIdx.x;
  const int blk = blockIdx.x;
  const int b   = blk >> 3;                        // NCHUNK == 8
  const int c0  = (blk & 7) * TB;

  for (int i = t; i < NJ*3; i += TB) offs[i] = offsets[i];
  __syncthreads();

  const int s = c0 + t;
  const float* pRow = pred + ((size_t)b * SDIM + s) * ROW;
  const float* gRow = tgt  + ((size_t)b * SDIM + s) * ROW;

  float a0=0.f,a1=0.f,a2=0.f,a3=0.f,a4=0.f,a5=0.f,a6=0.f,a7=0.f,a8=0.f;

  // ---- input-space terms: root (joint0 c<3, joint1 all), rot (joints 2..21)
  for (int i = 0; i < ROW; ++i) {
    float d = pRow[i] - gRow[i];
    float d2 = d * d;
    if (i < 3)                    a0 += d2;
    else if (i >= 6 && i < 12)    a1 += d2;
    else if (i >= 12)             a2 += d2;
  }

  // ---- tilt: b2 column of spine (joint 2) rotation
  {
    float Rp[9], Rg[9];
    d6_to_mat(pRow + 2*6, Rp);
    d6_to_mat(gRow + 2*6, Rg);
    float dx = Rp[3]-Rg[3], dy = Rp[4]-Rg[4], dz = Rp[5]-Rg[5];
    a8 += dx*dx + dy*dy + dz*dz;
  }

  // ---- FK for primary frame (pred then gt; sequential keeps registers low)
  fk_full(pRow, offs, &pos[t * POSSTRIDE]);
  fk_full(gRow, offs, &pos[t * POSSTRIDE + 66]);

  // ---- halo frames (+1, +2) for vel/accel at the chunk boundary
  if (t < 2 && (c0 + TB + t) < SDIM) {
    const int sh = c0 + TB + t;
    const int r  = TB + t;
    fk_full(pred + ((size_t)b * SDIM + sh) * ROW, offs, &pos[r * POSSTRIDE]);
    fk_full(tgt  + ((size_t)b * SDIM + sh) * ROW, offs, &pos[r * POSSTRIDE + 66]);
  }
  __syncthreads();

  // ---- cross-frame terms from LDS positions
  const bool hasV = (s + 1) < SDIM;
  const bool hasA = (s + 2) < SDIM;
  const float* r0 = &pos[t * POSSTRIDE];
  const float* r1 = &pos[(t + 1) * POSSTRIDE];
  const float* r2 = &pos[(t + 2) * POSSTRIDE];

  for (int j = 0; j < NJ; ++j) {
    const int o = j * 3;
    float p0x=r0[o], p0y=r0[o+1], p0z=r0[o+2];
    float g0x=r0[66+o], g0y=r0[66+o+1], g0z=r0[66+o+2];
    { float dx=p0x-g0x, dy=p0y-g0y, dz=p0z-g0z; a3 += dx*dx+dy*dy+dz*dz; }
    const bool foot = (j == 18) || (j == 21);
    if (foot) { float dy = p0y - g0y; a7 += dy*dy; }
    if (hasV) {
      float p1x=r1[o], p1y=r1[o+1], p1z=r1[o+2];
      float g1x=r1[66+o], g1y=r1[66+o+1], g1z=r1[66+o+2];
      float vpx=p1x-p0x, vpy=p1y-p0y, vpz=p1z-p0z;
      float vgx=g1x-g0x, vgy=g1y-g0y, vgz=g1z-g0z;
      { float dx=vpx-vgx, dy=vpy-vgy, dz=vpz-vgz; a4 += dx*dx+dy*dy+dz*dz; }
      if (foot) {
        float ng = sqrtf(vgx*vgx + vgy*vgy + vgz*vgz);
        if (ng < 0.005f) a6 += sqrtf(vpx*vpx + vpy*vpy + vpz*vpz);
      }
      if (hasA) {
        float p2x=r2[o], p2y=r2[o+1], p2z=r2[o+2];
        float g2x=r2[66+o], g2y=r2[66+o+1], g2z=r2[66+o+2];
        float apx=(p2x-p1x)-vpx, apy=(p2y-p1y)-vpy, apz=(p2z-p1z)-vpz;
        float agx=(g2x-g1x)-vgx, agy=(g2y-g1y)-vgy, agz=(g2z-g1z)-vgz;
        float dx=apx-agx, dy=apy-agy, dz=apz-agz;
        a5 += dx*dx + dy*dy + dz*dz;
      }
    }
  }

  // ---- stage per-thread partials (16 rows; 9 used) for the matrix-pipe reduce
  accL[t*ACCSTRIDE+0]=a0; accL[t*ACCSTRIDE+1]=a1; accL[t*ACCSTRIDE+2]=a2;
  accL[t*ACCSTRIDE+3]=a3; accL[t*ACCSTRIDE+4]=a4; accL[t*ACCSTRIDE+5]=a5;
  accL[t*ACCSTRIDE+6]=a6; accL[t*ACCSTRIDE+7]=a7; accL[t*ACCSTRIDE+8]=a8;
#pragma unroll
  for (int m = 9; m < 16; ++m) accL[t*ACCSTRIDE+m] = 0.f;
  __syncthreads();

  // ---- WMMA reduction: D(16x16) += A(16x4) * ones(4x16), 16 steps cover 64
  // threads. Row m of D = block total of accumulator m (f32, exact same math
  // as a scalar tree). Both waves compute identical results; no divergence at
  // the WMMA (EXEC all ones).
  const int lane = t & 31;
  const int mrow = lane & 15;              // A row (accumulator index)
  const int kb   = (lane >> 4) << 1;       // K base: lanes 0-15 -> K0/1, 16-31 -> K2/3
  v2f bOnes; bOnes.x = 1.0f; bOnes.y = 1.0f;
  v8f dAcc = {};
#pragma unroll
  for (int i = 0; i < 16; ++i) {
    v2f aFrag;
    aFrag.x = accL[(i*4 + kb    ) * ACCSTRIDE + mrow];
    aFrag.y = accL[(i*4 + kb + 1) * ACCSTRIDE + mrow];
    dAcc = __builtin_amdgcn_wmma_f32_16x16x4_f32(false, aFrag, false, bOnes,
                                                 (short)0, dAcc, false, false);
  }
  // D layout: lanes 0-15 VGPR r = row r; lanes 16-31 VGPR r = row r+8.
  if (t == 0) {
#pragma unroll
    for (int m = 0; m < 8; ++m) partials[(size_t)blk * PSTRIDE + m] = dAcc[m];
  }
  if (t == 16) partials[(size_t)blk * PSTRIDE + 8] = dAcc[0];
}

__global__ __launch_bounds__(64) void motion_loss_final(
    const float* __restrict__ partials, float* __restrict__ out) {
  __shared__ double red[9 * 64];
  const int t = threadIdx.x;
  double s[9];
#pragma unroll
  for (int m = 0; m < 9; ++m) s[m] = 0.0;
  for (int blk = t; blk < NBLK; blk += 64) {
#pragma unroll
    for (int m = 0; m < 9; ++m) s[m] += (double)partials[(size_t)blk * PSTRIDE + m];
  }
#pragma unroll
  for (int m = 0; m < 9; ++m) red[m*64 + t] = s[m];
  __syncthreads();
  if (t == 0) {
    double tot[9];
    for (int m = 0; m < 9; ++m) {
      double acc = 0.0;
      for (int i = 0; i < 64; ++i) acc += red[m*64 + i];
      tot[m] = acc;
    }
    const double BS = (double)BDIM * (double)SDIM;
    double l_root    = tot[0] / (BS * 3.0) + tot[1] / (BS * 6.0);
    double l_rot     = tot[2] / (BS * 120.0);
    double l_pos     = tot[3] / (BS * 66.0);
    double l_vel     = tot[4] / ((double)BDIM * (SDIM - 1) * 66.0);
    double l_smooth  = tot[5] / ((double)BDIM * (SDIM - 2) * 66.0);
    double l_contact = tot[6] / ((double)BDIM * (SDIM - 1) * 2.0);
    double l_floor   = tot[7] / (BS * 2.0);
    double l_tilt    = tot[8] / (BS * 3.0);
    double total = 5.0*l_root + 1.0*l_rot + 2.0*l_pos + 1.0*l_vel
                 + 0.5*l_smooth + 2.0*l_contact + 2.0*l_floor + 1.0*l_tilt;
    out[0] = (float)total;
  }
}

extern "C" void kernel_launch(void* const* d_in, const int* in_sizes, int n_in,
                              void* d_out, int out_size, void* d_ws, size_t ws_size,
                              hipStream_t stream) {
  const float* pred    = (const float*)d_in[0];
  const float* tgt     = (const float*)d_in[1];
  const float* offsets = (const float*)d_in[2];
  float* partials = (float*)d_ws;   // 2048 * 16 * 4 B = 128 KB, slots 0..8 of each block rewritten each call
  float* out = (float*)d_out;

  motion_loss_main<<<NBLK, TB, 0, stream>>>(pred, tgt, offsets, partials);
  motion_loss_final<<<1, 64, 0, stream>>>(partials, out);
  (void)in_sizes; (void)n_in; (void)out_size; (void)ws_size;
}